// embedding_num_past_variables_20727512171029
// MI455X (gfx1250) — compile-verified
//
#include <hip/hip_runtime.h>

typedef __attribute__((ext_vector_type(2))) float v2f;
typedef __attribute__((ext_vector_type(8))) float v8f;

// Problem dims (fixed by the reference)
#define Bn 32
#define Ln 512
#define Cn 32
#define Dn 512

// out[b,l,c,d] = x[b,l,c] * W[c,d] + bias[c,d]
// One wave = one (b, c, l_tile of 16); loops over 32 d-tiles of 16.
// Each tile is one v_wmma_f32_16x16x4_f32 outer-product (x in K=0 of A,
// W in K=0 of B, bias broadcast into C) -> exact f32 fma semantics.
__global__ __launch_bounds__(256) void embed_outer_wmma_kernel(
    const float* __restrict__ x,     // [B, L, C]
    const float* __restrict__ W,     // [C, D]
    const float* __restrict__ bias,  // [C, D]
    float* __restrict__ out)         // [B, L, C, D]
{
    const int tid  = blockIdx.x * blockDim.x + threadIdx.x;
    const int wave = tid >> 5;
    const int lane = threadIdx.x & 31;

    // wave -> (b, c, ltile): 32 * 32 * 32 = 32768 waves total
    const int ltile = wave & 31;
    const int c     = (wave >> 5) & 31;
    const int b     = wave >> 10;

    const int l_base   = ltile * 16;
    const int idx16    = lane & 15;
    const bool lowHalf = (lane < 16);

    // ---- A matrix (16x4 f32, 2 VGPRs): VGPR0 lanes0-15 = K=0 column = x[l] ----
    // Load for all lanes (same 16 values twice, same cache lines), mask into K=0.
    const float xv = x[(size_t)(b * Ln + l_base + idx16) * Cn + c];
    v2f a;
    a.x = lowHalf ? xv : 0.0f;   // K=0 (lanes 0-15); K=2 (lanes 16-31) = 0
    a.y = 0.0f;                  // K=1 / K=3 = 0

    const size_t wrow = (size_t)c * Dn;

    // Per-lane output element index: row = l_base + (upper half -> +8), col = idx16.
    const int rowoff   = lowHalf ? 0 : 8;
    const size_t obase = ((size_t)(b * Ln + l_base + rowoff) * Cn + c) * (size_t)Dn + idx16;

    for (int dt = 0; dt < Dn / 16; ++dt) {
        const int d_base = dt * 16;

        // W / bias tile for this lane's column (hot in L2: only 128 KB total)
        const float wv = W[wrow + d_base + idx16];
        const float bv = bias[wrow + d_base + idx16];

        // ---- B matrix (4x16 f32, 2 VGPRs): VGPR0 lanes0-15 = K=0 row = W[d] ----
        v2f bm;
        bm.x = lowHalf ? wv : 0.0f;
        bm.y = 0.0f;

        // ---- C matrix (16x16 f32, 8 VGPRs): bias is row-independent -> broadcast ----
        v8f cm = {bv, bv, bv, bv, bv, bv, bv, bv};

        // D = A x B + C  ==  x[l] * W[d] + bias[d]  (exact: all other K terms are 0*0)
        v8f d = __builtin_amdgcn_wmma_f32_16x16x4_f32(
            /*neg_a=*/false, a, /*neg_b=*/false, bm,
            /*c_mod=*/(short)0, cm, /*reuse_a=*/false, /*reuse_b=*/false);

        // ---- Store: VGPR r -> row l_base + r (+8 upper half), col d_base + idx16.
        // Output is 1 GiB, write-once: non-temporal so it streams past the caches.
        float* o = out + obase + d_base;
        #pragma unroll
        for (int r = 0; r < 8; ++r) {
            __builtin_nontemporal_store(d[r], o + (size_t)r * (Cn * Dn));
        }
    }
}

extern "C" void kernel_launch(void* const* d_in, const int* in_sizes, int n_in,
                              void* d_out, int out_size, void* d_ws, size_t ws_size,
                              hipStream_t stream) {
    const float* x    = (const float*)d_in[0];  // [B, L, C] f32
    const float* W    = (const float*)d_in[1];  // [C, D]    f32
    const float* bias = (const float*)d_in[2];  // [C, D]    f32
    float* out        = (float*)d_out;          // [B, L, C, D] f32

    // 32768 waves, 8 waves (256 threads) per block -> 4096 blocks
    const int threads = 256;
    const int total_waves = Bn * Cn * (Ln / 16);
    const int blocks = (total_waves * 32) / threads;

    embed_outer_wmma_kernel<<<blocks, threads, 0, stream>>>(x, W, bias, out);
}